// QuantumExchangeabilityTest_25872882991128
// MI455X (gfx1250) — compile-verified
//
#include <hip/hip_runtime.h>
#include <hip/hip_bf16.h>

typedef __attribute__((ext_vector_type(2))) float v2f;
typedef __attribute__((ext_vector_type(8))) float v8f;

#define NPTS 320
#define DIM  64

// ---------------------------------------------------------------------------
// Kernel 1: G0 = X * X^T  (320x320, K=64) via V_WMMA_F32_16X16X4_F32.
// One wave per 16x16 output tile; 16 WMMA ops per tile (K stepped by 4).
// A layout (16x4 f32): lane 0-15 -> M=lane, VGPR0=K0,VGPR1=K1;
//                      lane 16-31 -> M=lane-16, VGPR0=K2,VGPR1=K3.
// B layout mirrors A with N in lanes. C/D: VGPR r -> M = r + 8*(lane>>4),
// N = lane&15.
// ---------------------------------------------------------------------------
__global__ __launch_bounds__(32)
void gram_wmma_kernel(const float* __restrict__ X, float* __restrict__ G)
{
    const int tm   = blockIdx.x;        // row tile   (0..19)
    const int tn   = blockIdx.y;        // col tile   (0..19)
    const int lane = threadIdx.x;       // 0..31, EXEC all ones (WMMA req.)
    const int l15  = lane & 15;
    const int kh   = lane >> 4;         // which K-half this lane holds

    const float* __restrict__ Arow = X + (tm * 16 + l15) * DIM;
    const float* __restrict__ Brow = X + (tn * 16 + l15) * DIM;

    v8f c = {0.f, 0.f, 0.f, 0.f, 0.f, 0.f, 0.f, 0.f};

#pragma unroll
    for (int k0 = 0; k0 < DIM; k0 += 4) {
        v2f a, b;
        a.x = Arow[k0 + 2 * kh + 0];
        a.y = Arow[k0 + 2 * kh + 1];
        b.x = Brow[k0 + 2 * kh + 0];
        b.y = Brow[k0 + 2 * kh + 1];
        // 8 args: (neg_a, A, neg_b, B, c_mod, C, reuse_a, reuse_b)
        c = __builtin_amdgcn_wmma_f32_16x16x4_f32(
                false, a, false, b, (short)0, c, false, false);
    }

#pragma unroll
    for (int r = 0; r < 8; ++r) {
        const int mm = tm * 16 + r + 8 * kh;
        G[mm * NPTS + tn * 16 + l15] = c[r];
    }
}

// ---------------------------------------------------------------------------
// Kernel 2: per-configuration distance-variance using the precomputed G0.
// Block p (0 = identity, 1..P = perms), 320 threads (10 wave32s).
//   sq[p][i][j] = s_i^2*sn0[si] + s_j^2*sn0[sj] - 2*s_i*s_j*G0[si,sj]
//   v = F[i][j] * sqrt(max(sq,0));  var(ddof=1) over i<j pairs.
// Ordered-pair sums are exactly 2x the unordered sums (v symmetric), and the
// j==i term is exactly 0, so the inner loop runs branch-free over all j.
// Deterministic shuffle+LDS reduction (no float atomics).
// ---------------------------------------------------------------------------
__global__ __launch_bounds__(NPTS)
void perm_var_kernel(const float* __restrict__ G,
                     const int*   __restrict__ types,
                     const int*   __restrict__ perms,
                     const float* __restrict__ fermionp,
                     const float* __restrict__ bosonp,
                     float* __restrict__ out)
{
    __shared__ int   sSig[NPTS];
    __shared__ float sS  [NPTS];
    __shared__ float sE  [NPTS];
    __shared__ int   sT  [NPTS];
    __shared__ float wsum[NPTS / 32];
    __shared__ float wsq [NPTS / 32];

    const int p = blockIdx.x;
    const int i = threadIdx.x;

    const float fa = fermionp[0];
    const float bs = bosonp[0];

    const int sig = (p == 0) ? i : perms[(p - 1) * NPTS + i];
    const int ti  = types[i];

    float s = 1.0f;
    if (p != 0 && sig != i) {
        const int ts = types[sig];
        s = (ti == 0 && ts == 0) ? fa : ((ti == 1 && ts == 1) ? bs : 1.0f);
    }
    const float sn = G[sig * NPTS + sig];   // |data[sig]|^2 from G0 diagonal
    const float ei = s * s * sn;

    sSig[i] = sig;
    sS[i]   = s;
    sE[i]   = ei;
    sT[i]   = ti;
    __syncthreads();

    // F[i][j] as a function of t_j for this fixed i
    const float f_tj0 = (ti == 0) ? fa : 1.0f;
    const float f_tj1 = (ti == 1) ? bs : 1.0f;

    const float* __restrict__ Grow = G + sig * NPTS;
    const float si2 = 2.0f * s;

    float sum = 0.0f, sumsq = 0.0f;
    for (int j = 0; j < NPTS; ++j) {
        const float g = Grow[sSig[j]];               // L2-resident gather
        float d = ei + sE[j] - si2 * sS[j] * g;
        d = fmaxf(d, 0.0f);
        const float F = sT[j] ? f_tj1 : f_tj0;
        const float v = F * __builtin_sqrtf(d);
        sum   += v;
        sumsq += v * v;
    }

    // intra-wave reduction (wave32)
#pragma unroll
    for (int off = 16; off > 0; off >>= 1) {
        sum   += __shfl_down(sum,   off, 32);
        sumsq += __shfl_down(sumsq, off, 32);
    }
    if ((i & 31) == 0) {
        wsum[i >> 5] = sum;
        wsq [i >> 5] = sumsq;
    }
    __syncthreads();

    if (i == 0) {
        double S = 0.0, S2 = 0.0;
        for (int w = 0; w < NPTS / 32; ++w) { S += wsum[w]; S2 += wsq[w]; }
        S  *= 0.5;                       // ordered -> unordered pairs
        S2 *= 0.5;
        const double M = (double)NPTS * (NPTS - 1) / 2.0;   // 51040
        out[p] = (float)((S2 - S * S / M) / (M - 1.0));
    }
}

extern "C" void kernel_launch(void* const* d_in, const int* in_sizes, int n_in,
                              void* d_out, int out_size, void* d_ws, size_t ws_size,
                              hipStream_t stream)
{
    const float* data  = (const float*)d_in[0];   // (320, 64) f32
    const float* fa    = (const float*)d_in[1];   // scalar f32
    const float* bs    = (const float*)d_in[2];   // scalar f32
    const int*   types = (const int*)  d_in[3];   // (320,) i32
    const int*   perms = (const int*)  d_in[4];   // (1000, 320) i32
    float*       out   = (float*)d_out;           // (1001,) f32
    float*       G     = (float*)d_ws;            // 320*320*4 = 409600 B scratch

    // 1) one f32-WMMA GEMM: G0 = data * data^T
    gram_wmma_kernel<<<dim3(NPTS / 16, NPTS / 16), 32, 0, stream>>>(data, G);

    // 2) per-configuration variance (out_size = P+1 = 1001 blocks)
    perm_var_kernel<<<out_size, NPTS, 0, stream>>>(G, types, perms, fa, bs, out);
}